// HSMNet_9096740733333
// MI455X (gfx1250) — compile-verified
//
#include <hip/hip_runtime.h>
#include <hip/hip_bf16.h>
#include <stdint.h>

#define DEVINLINE __device__ __forceinline__

typedef float v4f __attribute__((ext_vector_type(4)));

// On gfx1250 the flat (generic) address of an LDS object carries the LDS byte
// offset in its low 32 bits (aperture id lives in addr[63:32] per the ISA
// aperture rules), so truncation yields the DS-space byte address the async
// DMA path wants in its VDST VGPR.
DEVINLINE uint32_t lds_byte_addr(const void* p) {
  return (uint32_t)(uintptr_t)p;
}

// CDNA5 async global->LDS loads (ASYNCcnt-tracked), CDNA5 ISA.
DEVINLINE void async_load_lds_b128(uint32_t lds, const void* gaddr) {
  asm volatile("global_load_async_to_lds_b128 %0, %1, off"
               :: "v"(lds), "v"(gaddr) : "memory");
}
DEVINLINE void wait_async0() {
  asm volatile("s_wait_asynccnt 0x0" ::: "memory");
}

// cost[b,c,d,h,w] = |ref[b,c,h,w] - tgt[b,c,h,w-d]| for w >= d else 0.
//
// Fixed 64-lane blocks (2 waves). Each lane owns 4 consecutive w; a block
// covers ROWS rows so (W/4)*ROWS == 64 for every pyramid level. tgt row slabs
// are staged with one async b128 DMA per lane; ref comes in as an aligned
// float4; each d-iteration emits a single b128 non-temporal store (write-once
// data must not displace the 192MB L2's useful lines).
template <int C, int D, int H, int W, int ROWS>
__global__ void __launch_bounds__(64) cost_volume_kernel(
    const float* __restrict__ ref, const float* __restrict__ tgt,
    float* __restrict__ out) {
  constexpr int LPR = W / 4;               // lanes per row
  static_assert(LPR * ROWS == 64, "block shape");
  __shared__ __align__(16) float s_tgt[ROWS * W];   // 1 KB

  const int t   = threadIdx.x;
  const int sub = t / LPR;                 // row within block
  const int w0  = (t % LPR) * 4;           // first of 4 owned columns
  const int rid = blockIdx.x * ROWS + sub; // flattened (b*C + c)*H + h
  const int h   = rid % H;                 // compile-time H -> mul/shift
  const int bc  = rid / H;

  async_load_lds_b128(lds_byte_addr(&s_tgt[sub * W + w0]), tgt + rid * W + w0);
  const v4f rv = *(const v4f*)(ref + rid * W + w0);
  wait_async0();
  __syncthreads();

  constexpr long long plane = (long long)H * W;
  long long idx = (long long)bc * D * plane + (long long)h * W + w0;
  const float* srow = &s_tgt[sub * W];
  #pragma unroll 4
  for (int d = 0; d < D; ++d, idx += plane) {
    v4f v;
    #pragma unroll
    for (int i = 0; i < 4; ++i) {
      const int w = w0 + i;
      float x = 0.0f;
      if (w >= d) x = fabsf(rv[i] - srow[w - d]);
      v[i] = x;
    }
    __builtin_nontemporal_store(v, (v4f*)(out + idx));  // th:TH_STORE_NT b128
  }
}

// Fused finest-level head: cost3 = -sum_c |conv10 - conv11(shifted)|, softmax
// over 96 disparity bins, regression with disp = 8*d — computed straight from
// the 12 MB inputs (async-DMA'd 16x256 row slabs in LDS) instead of re-reading
// the 604 MB feat3 volume. Masked bins (w < d) carry logit exactly 0 (not
// -inf), matching the reference's 0*valid element.
__global__ void __launch_bounds__(256) pred3_kernel(
    const float* __restrict__ ref, const float* __restrict__ tgt,
    float* __restrict__ pred) {
  constexpr int Cc = 16, Hh = 192, Ww = 256, Dd = 96;
  __shared__ __align__(16) float sR[Cc * Ww];   // 16 KB
  __shared__ __align__(16) float sT[Cc * Ww];   // 16 KB
  const int t  = threadIdx.x;     // 256 threads = 8 waves
  const int bh = blockIdx.x;      // b*H + h
  const long long rowbase =
      (long long)(bh / Hh) * Cc * Hh * Ww + (long long)(bh % Hh) * Ww;

  #pragma unroll
  for (int k = 0; k < 4; ++k) {
    int i4 = t + k * 256;                  // 1024 float4 per slab
    int c = i4 >> 6, w4 = (i4 & 63) << 2;
    long long g = rowbase + (long long)c * (Hh * Ww) + w4;
    async_load_lds_b128(lds_byte_addr(&sR[c * Ww + w4]), ref + g);
    async_load_lds_b128(lds_byte_addr(&sT[c * Ww + w4]), tgt + g);
  }
  wait_async0();
  __syncthreads();

  const int w = t;
  float rv[Cc];
  #pragma unroll
  for (int c = 0; c < Cc; ++c) rv[c] = sR[c * Ww + w];

  // Single-pass online softmax + regression accumulate.
  float m = -3.402823466e+38f, l = 0.0f, acc = 0.0f;
  for (int d = 0; d < Dd; ++d) {
    float s = 0.0f;
    if (w >= d) {
      float sum = 0.0f;
      #pragma unroll
      for (int c = 0; c < Cc; ++c) sum += fabsf(rv[c] - sT[c * Ww + (w - d)]);
      s = -sum;
    }
    float mn = fmaxf(m, s);
    float scale = __expf(m - mn);
    float e = __expf(s - mn);
    l   = l   * scale + e;
    acc = acc * scale + e * (float)(8 * d);
    m = mn;
  }
  pred[(long long)bh * Ww + w] = acc / l;
}

extern "C" void kernel_launch(void* const* d_in, const int* in_sizes, int n_in,
                              void* d_out, int out_size, void* d_ws, size_t ws_size,
                              hipStream_t stream) {
  (void)in_sizes; (void)n_in; (void)out_size; (void)d_ws; (void)ws_size;
  const float* c40 = (const float*)d_in[0];
  const float* c41 = (const float*)d_in[1];
  const float* c30 = (const float*)d_in[2];
  const float* c31 = (const float*)d_in[3];
  const float* c20 = (const float*)d_in[4];
  const float* c21 = (const float*)d_in[5];
  const float* c10 = (const float*)d_in[6];
  const float* c11 = (const float*)d_in[7];
  // d_in[8] = maxdisp (=768 per setup_inputs); D per level is shape-implied.

  float* out  = (float*)d_out;
  float* pred = out;                 // [2,192,256]            :      98304
  float* f6   = out + 98304;         // [2,32,12,24,32]        :     589824
  float* f5   = out + 688128;        // [2,16,24,48,64]        :    2359296
  float* f4   = out + 3047424;       // [2,16,48,96,128]       :   18874368
  float* f3   = out + 21921792;      // [2,16,96,192,256]      :  150994944

  // grid = B*C*H / ROWS, 64 threads per block (2 waves, wave32)
  cost_volume_kernel<32, 12, 24,  32,  8><<<192,  64, 0, stream>>>(c40, c41, f6);
  cost_volume_kernel<16, 24, 48,  64,  4><<<384,  64, 0, stream>>>(c30, c31, f5);
  cost_volume_kernel<16, 48, 96,  128, 2><<<1536, 64, 0, stream>>>(c20, c21, f4);
  cost_volume_kernel<16, 96, 192, 256, 1><<<6144, 64, 0, stream>>>(c10, c11, f3);
  // blocks = B*H, 256 threads
  pred3_kernel<<<2 * 192, 256, 0, stream>>>(c10, c11, pred);
}